// LULayer_33569464386190
// MI455X (gfx1250) — compile-verified
//
#include <hip/hip_runtime.h>
#include <hip/hip_bf16.h>

typedef __attribute__((ext_vector_type(2))) float v2f;
typedef __attribute__((ext_vector_type(4))) float v4f;
typedef __attribute__((ext_vector_type(8))) float v8f;

#define DIM     256
#define BATCH   131072
#define NSTRIPS (BATCH / 16)      // 8192 strips of 16 rows
#define NWG     512
#define TPB     256               // 8 wave32 waves per workgroup
#define WPW     (TPB / 32)

// ---------------------------------------------------------------------------
// Prep kernel 1: recover the permutation from the dense permutation matrix p.
// perm[j] = s such that p[j][s] == 1.
// ---------------------------------------------------------------------------
__global__ void k_perm(const float* __restrict__ p, int* __restrict__ perm) {
    int j = blockIdx.x * blockDim.x + threadIdx.x;
    if (j >= DIM) return;
    int idx = 0;
    for (int s = 0; s < DIM; ++s)
        if (p[j * DIM + s] > 0.5f) idx = s;
    perm[j] = idx;
}

// ---------------------------------------------------------------------------
// Prep kernel 2: B[k][n] = w[n][k] = sum_t l[perm[n]][t] * u[t][k]
// Stored in PAIRED layout so a WMMA B-fragment is one contiguous float2:
//   wPair[((k>>1)*DIM + n)*2 + (k&1)] = B[k][n]        (k even/odd interleave)
// ---------------------------------------------------------------------------
__global__ void k_wt(const float* __restrict__ l, const float* __restrict__ u,
                     const int* __restrict__ perm, float* __restrict__ wPair) {
    int tid = blockIdx.x * blockDim.x + threadIdx.x;   // 65536 threads
    int n = tid & (DIM - 1);
    int k = tid >> 8;
    const float* lrow = l + (size_t)perm[n] * DIM;
    float acc = 0.0f;
    for (int t = 0; t < DIM; ++t)
        acc = fmaf(lrow[t], u[t * DIM + k], acc);
    wPair[((size_t)(k >> 1) * DIM + n) * 2 + (k & 1)] = acc;
}

// ---------------------------------------------------------------------------
// Prep kernel 3: log|det| = sum_k log|u[k][k]|   (unit-triangular -> 0.0)
// ---------------------------------------------------------------------------
__global__ void k_logdet(const float* __restrict__ u, float* __restrict__ out) {
    __shared__ float buf[DIM];
    int t = threadIdx.x;
    buf[t] = __logf(fabsf(u[t * DIM + t]));
    __syncthreads();
    for (int s = DIM / 2; s > 0; s >>= 1) {
        if (t < s) buf[t] += buf[t + s];
        __syncthreads();
    }
    if (t == 0) out[(size_t)BATCH * DIM] = buf[0];
}

// ---------------------------------------------------------------------------
// Main GEMM: y[m][n] = sum_k x[m][k] * B[k][n],  B resident in LDS (256 KB).
// Each wave: one 16-row strip of x -> all 256 output columns, so x streams
// from HBM exactly once. Inner math: V_WMMA_F32_16X16X4_F32.
// ---------------------------------------------------------------------------
__global__ void __launch_bounds__(TPB, 1)
k_gemm(const float* __restrict__ x, const float* __restrict__ wPair,
       float* __restrict__ y) {
    extern __shared__ float lds[];                 // 256 KB: paired B matrix

    // Cooperative load of B into LDS (coalesced b128).
    for (int i = threadIdx.x; i < DIM * DIM / 4; i += TPB)
        ((v4f*)lds)[i] = ((const v4f*)wPair)[i];
    __syncthreads();

    const int lane = threadIdx.x & 31;
    const int wave = threadIdx.x >> 5;
    const int lo   = lane & 15;                    // M index within A / N within B
    const int hi   = lane >> 4;                    // selects K pair (0,1) vs (2,3)
    const int nwaves = gridDim.x * WPW;
    const int wgid   = blockIdx.x * WPW + wave;

    const v2f* lds2 = (const v2f*)lds;             // lds2[(k>>1)*DIM + n] = {B[k][n], B[k+1][n]}

    for (int strip = wgid; strip < NSTRIPS; strip += nwaves) {
        const int row0 = strip * 16;

        // ---- Load A block (16 rows x 256 K) into 128 VGPRs in WMMA order.
        // lane<16: row lo, cols {4k,4k+1}; lane>=16: row lo, cols {4k+2,4k+3}
        const float* arow = x + (size_t)(row0 + lo) * DIM + 2 * hi;
        v2f a[64];
#pragma unroll
        for (int kk = 0; kk < 64; ++kk)
            a[kk] = *(const v2f*)(arow + 4 * kk);

        // Prefetch the next strip's rows while we compute this one.
        {
            int next = strip + nwaves;
            if (next < NSTRIPS)
                __builtin_prefetch(x + (size_t)(next * 16 + lo) * DIM + 2 * hi, 0, 0);
        }

        // ---- 16 N-tiles of 16 columns; 64 K-steps each => 1024 WMMAs/strip.
#pragma unroll 1
        for (int nt = 0; nt < 16; ++nt) {
            const int n = nt * 16 + lo;
            v8f c = {};
#pragma unroll
            for (int kk = 0; kk < 64; ++kk) {
                // B fragment: lane<16 holds K=4kk+{0,1}; lane>=16 K=4kk+{2,3}
                v2f b = lds2[(size_t)(2 * kk + hi) * DIM + n];
                c = __builtin_amdgcn_wmma_f32_16x16x4_f32(
                        /*neg_a=*/false, a[kk],
                        /*neg_b=*/false, b,
                        /*c_mod=*/(short)0, c,
                        /*reuse_a=*/false, /*reuse_b=*/false);
            }
            // ---- Store C: VGPR v -> rows (v | v+8), col n.
            float* yb = y + (size_t)row0 * DIM + n;
#pragma unroll
            for (int v = 0; v < 8; ++v)
                yb[(size_t)(v + 8 * hi) * DIM] = c[v];
        }
    }
}

// ---------------------------------------------------------------------------
extern "C" void kernel_launch(void* const* d_in, const int* in_sizes, int n_in,
                              void* d_out, int out_size, void* d_ws, size_t ws_size,
                              hipStream_t stream) {
    const float* x = (const float*)d_in[0];
    const float* p = (const float*)d_in[1];
    const float* l = (const float*)d_in[2];
    const float* u = (const float*)d_in[3];
    float* y = (float*)d_out;

    int*   perm  = (int*)d_ws;                          // 1 KB
    float* wPair = (float*)((char*)d_ws + 1024);        // 256 KB paired B

    k_perm  <<<1, DIM, 0, stream>>>(p, perm);
    k_wt    <<<DIM * DIM / 256, 256, 0, stream>>>(l, u, perm, wPair);
    k_logdet<<<1, DIM, 0, stream>>>(u, y);
    k_gemm  <<<NWG, TPB, DIM * DIM * sizeof(float), stream>>>(x, wPair, y);
}